// MultiHeadAttention_28028956574019
// MI455X (gfx1250) — compile-verified
//
#include <hip/hip_runtime.h>
#include <math.h>

// ---------------------------------------------------------------------------
// MI455X (gfx1250, wave32) implementation of windowed multi-head attention.
// Stages:
//   A) qkv = query @ qkv_w^T + qkv_b        (WMMA f32 16x16x4 GEMM)
//   B) dx  = sigmoid(relu(query)@off_w^T@lin2_w^T + b) * L   (VALU, tiny)
//   C) per (b, group, 16-row q tile): flash-attention over future window,
//      S and P@V on the f32 WMMA pipe, online softmax, streaming output.
// ---------------------------------------------------------------------------

typedef float  v2f __attribute__((ext_vector_type(2)));
typedef float  v8f __attribute__((ext_vector_type(8)));

#define BATCH 8
#define SEQ   2048
#define HID   1024
#define QKVC  3072          // 3*HID
#define GROUPS 4
#define DHEAD 256           // HID / GROUPS
#define D1    256

__device__ __forceinline__ v8f wmma_f32(v2f a, v2f b, v8f c) {
    // V_WMMA_F32_16X16X4_F32 : D = A(16x4) * B(4x16) + C(16x16), fp32
    return __builtin_amdgcn_wmma_f32_16x16x4_f32(false, a, false, b,
                                                 (short)0, c, false, false);
}

// ---------------------------------------------------------------------------
// Kernel A: C[16384,3072] = X[16384,1024] @ W[3072,1024]^T + bias
// block = 256 threads (8 waves); block tile 128(M) x 64(N); K-panel 64.
// ---------------------------------------------------------------------------
#define A_STRIDE 68   // 64 + 4 pad: float4-aligned, bank pattern (4r+k)%64
__global__ __launch_bounds__(256) void qkv_gemm_kernel(
    const float* __restrict__ X, const float* __restrict__ W,
    const float* __restrict__ bias, float* __restrict__ C)
{
    __shared__ float a_lds[128 * A_STRIDE];
    __shared__ float b_lds[64 * A_STRIDE];

    const int tid  = threadIdx.x;
    const int lane = tid & 31;
    const int wave = tid >> 5;
    const int lo   = lane & 15;
    const int hi   = lane >> 4;

    const int rbase = blockIdx.y * 128;
    const int cbase = blockIdx.x * 64;

    v8f acc[4];
#pragma unroll
    for (int i = 0; i < 4; ++i)
#pragma unroll
        for (int e = 0; e < 8; ++e) acc[i][e] = 0.0f;

    for (int k0 = 0; k0 < HID; k0 += 64) {
        __syncthreads();
        {   // A tile: 128 rows x 64 k
            const int row = tid >> 1;
            const int c0  = (tid & 1) * 32;
            const float* src = X + (size_t)(rbase + row) * HID + k0 + c0;
            float* dst = &a_lds[row * A_STRIDE + c0];
#pragma unroll
            for (int i = 0; i < 8; ++i)
                *(float4*)(dst + i * 4) = *(const float4*)(src + i * 4);
        }
        {   // B tile: 64 rows (output cols) x 64 k
            const int row = tid >> 2;
            const int c0  = (tid & 3) * 16;
            const float* src = W + (size_t)(cbase + row) * HID + k0 + c0;
            float* dst = &b_lds[row * A_STRIDE + c0];
#pragma unroll
            for (int i = 0; i < 4; ++i)
                *(float4*)(dst + i * 4) = *(const float4*)(src + i * 4);
        }
        __syncthreads();

        const float* arow = &a_lds[(wave * 16 + lo) * A_STRIDE + hi * 2];
#pragma unroll
        for (int kk = 0; kk < 16; ++kk) {
            v2f a = *(const v2f*)(arow + kk * 4);
#pragma unroll
            for (int nt = 0; nt < 4; ++nt) {
                v2f b = *(const v2f*)&b_lds[(nt * 16 + lo) * A_STRIDE + kk * 4 + hi * 2];
                acc[nt] = wmma_f32(a, b, acc[nt]);
            }
        }
    }

    // epilogue: C-layout -> VGPR v holds row (v + hi*8), col = lo
#pragma unroll
    for (int nt = 0; nt < 4; ++nt) {
        const int col = cbase + nt * 16 + lo;
        const float bv = bias[col];
        const int row0 = rbase + wave * 16 + hi * 8;
#pragma unroll
        for (int v = 0; v < 8; ++v)
            C[(size_t)(row0 + v) * QKVC + col] = acc[nt][v] + bv;
    }
}

// ---------------------------------------------------------------------------
// Kernel B: dx[row] = sigmoid(relu(x_row)@off_w^T @ lin2_w^T + lin2_b) * SEQ
// block = 256 threads handles 8 rows; thread j owns hidden unit j (D1==256).
// ---------------------------------------------------------------------------
__global__ __launch_bounds__(256) void offset_mlp_kernel(
    const float* __restrict__ X, const float* __restrict__ OW,
    const float* __restrict__ L2W, const float* __restrict__ L2B,
    float* __restrict__ DX)
{
    __shared__ float x_lds[8 * HID];
    __shared__ float red[256];

    const int tid   = threadIdx.x;
    const int rbase = blockIdx.x * 8;

    for (int i = tid; i < (8 * HID) / 4; i += 256) {
        float4 v = *(const float4*)(X + (size_t)rbase * HID + i * 4);
        v.x = fmaxf(v.x, 0.f); v.y = fmaxf(v.y, 0.f);
        v.z = fmaxf(v.z, 0.f); v.w = fmaxf(v.w, 0.f);
        *(float4*)(&x_lds[i * 4]) = v;
    }
    __syncthreads();

    float acc[8];
#pragma unroll
    for (int r = 0; r < 8; ++r) acc[r] = 0.0f;

    const float* wrow = OW + (size_t)tid * HID;
    for (int k = 0; k < HID; k += 4) {
        const float4 w = *(const float4*)(wrow + k);
#pragma unroll
        for (int r = 0; r < 8; ++r) {
            const float* xr = &x_lds[r * HID + k];
            acc[r] = fmaf(w.x, xr[0], acc[r]);
            acc[r] = fmaf(w.y, xr[1], acc[r]);
            acc[r] = fmaf(w.z, xr[2], acc[r]);
            acc[r] = fmaf(w.w, xr[3], acc[r]);
        }
    }
    const float lw = L2W[tid];
#pragma unroll
    for (int r = 0; r < 8; ++r) acc[r] *= lw;

    for (int r = 0; r < 8; ++r) {
        red[tid] = acc[r];
        __syncthreads();
        for (int s = 128; s > 0; s >>= 1) {
            if (tid < s) red[tid] += red[tid + s];
            __syncthreads();
        }
        if (tid == 0) {
            const float z = red[0] + L2B[0];
            DX[rbase + r] = (1.0f / (1.0f + __expf(-z))) * (float)SEQ;
        }
        __syncthreads();
    }
}

// ---------------------------------------------------------------------------
// Kernel C: flash attention per (b, group, 16-query-row tile).
// block = 128 threads = 4 waves; each wave owns a 64-wide slice of d.
// ---------------------------------------------------------------------------
#define KSTRIDE 260   // 256 + 4 pad (float4 aligned, banks (4r+k)%64)
#define VSTRIDE 18    // v stored transposed [d][key], even pad for float2
#define PSTRIDE 18

__global__ __launch_bounds__(128) void attention_kernel(
    const float* __restrict__ QKV, const float* __restrict__ DX,
    float* __restrict__ OUT)
{
    __shared__ float k_lds[16 * KSTRIDE];
    __shared__ float v_lds[DHEAD * VSTRIDE];
    __shared__ float s_lds[4 * 256];
    __shared__ float p_lds[4 * 16 * PSTRIDE];

    const int tid  = threadIdx.x;
    const int lane = tid & 31;
    const int wave = tid >> 5;
    const int lo   = lane & 15;
    const int hi   = lane >> 4;

    const int qt = blockIdx.x;                 // 0..127 (16 rows each)
    const int g  = blockIdx.y;                 // 0..3
    const int b  = blockIdx.z;                 // 0..7
    const int l0 = qt * 16;
    const float wsz    = (float)(4 << (2 * g));          // 4,16,64,256
    const float scale2 = 2.0f / sqrtf((float)SEQ);       // doubled logits

    const size_t base = (size_t)b * SEQ * QKVC;

    // per-row window width dx (incl. window size), replicated per C-layout row
    const float dq = DX[b * SEQ + l0 + lo] + wsz;
    float dxv[8];
#pragma unroll
    for (int v = 0; v < 8; ++v)
        dxv[v] = __shfl(dq, v + hi * 8, 32);
    float mdx = dq;
#pragma unroll
    for (int off = 16; off > 0; off >>= 1)
        mdx = fmaxf(mdx, __shfl_xor(mdx, off, 32));

    // Q fragments (A-layout): this wave's 64-d slice, 16 x (16x4) tiles
    const int dbase = g * DHEAD + wave * 64;
    v2f qa[16];
    {
        const float* qrow = QKV + base + (size_t)(l0 + lo) * QKVC + dbase + hi * 2;
#pragma unroll
        for (int s = 0; s < 16; ++s)
            qa[s] = *(const v2f*)(qrow + s * 4);
    }

    v8f o[4];
#pragma unroll
    for (int t = 0; t < 4; ++t)
#pragma unroll
        for (int e = 0; e < 8; ++e) o[t][e] = 0.0f;
    float rmax[8], rsum[8];
#pragma unroll
    for (int v = 0; v < 8; ++v) { rmax[v] = 0.0f; rsum[v] = 0.0f; }

    const int mt0 = qt;                                     // past keys masked
    int mt1 = (l0 + 15 + (int)mdx) >> 4;                    // dx-bounded future
    if (mt1 > SEQ / 16 - 1) mt1 = SEQ / 16 - 1;

    for (int mt = mt0; mt <= mt1; ++mt) {
        const int m0 = mt * 16;
        __syncthreads();   // protect LDS reuse from previous iteration

        {   // K tile: [key][d], stride KSTRIDE
            const int row = tid >> 3;
            const int c0  = (tid & 7) * 32;
            const float* src = QKV + base + (size_t)(m0 + row) * QKVC + HID + g * DHEAD + c0;
            float* dst = &k_lds[row * KSTRIDE + c0];
#pragma unroll
            for (int i = 0; i < 8; ++i)
                *(float4*)(dst + i * 4) = *(const float4*)(src + i * 4);
        }
        {   // V tile transposed: [d][key], stride VSTRIDE
            const int row = tid >> 3;
            const int c0  = (tid & 7) * 32;
            const float* src = QKV + base + (size_t)(m0 + row) * QKVC + 2 * HID + g * DHEAD + c0;
#pragma unroll
            for (int i = 0; i < 32; ++i)
                v_lds[(c0 + i) * VSTRIDE + row] = src[i];
        }
        __syncthreads();

        // partial S over this wave's 64-d slice: S_w = Q_w @ K_w^T
        v8f s;
#pragma unroll
        for (int e = 0; e < 8; ++e) s[e] = 0.0f;
        {
            const float* kb = &k_lds[lo * KSTRIDE + wave * 64 + hi * 2];
#pragma unroll
            for (int st = 0; st < 16; ++st) {
                v2f bf = *(const v2f*)(kb + st * 4);
                s = wmma_f32(qa[st], bf, s);
            }
        }

        // cross-wave reduction of partial S (layout-preserving exchange)
#pragma unroll
        for (int v = 0; v < 8; ++v)
            s_lds[wave * 256 + v * 32 + lane] = s[v];
        __syncthreads();

        float sf[8];
#pragma unroll
        for (int v = 0; v < 8; ++v)
            sf[v] = s_lds[v * 32 + lane]       + s_lds[256 + v * 32 + lane]
                  + s_lds[512 + v * 32 + lane] + s_lds[768 + v * 32 + lane];

        // mask + online softmax (C layout: row = l0 + v + hi*8, key = m0 + lo)
        const int m = m0 + lo;
#pragma unroll
        for (int v = 0; v < 8; ++v) {
            const int l = l0 + v + hi * 8;
            const bool valid = (m >= l) && ((float)(m - l) <= dxv[v]);
            const float z = valid ? sf[v] * scale2 : -1e30f;

            float t = z;                               // row max over 16 lanes
            t = fmaxf(t, __shfl_xor(t, 1, 32));
            t = fmaxf(t, __shfl_xor(t, 2, 32));
            t = fmaxf(t, __shfl_xor(t, 4, 32));
            t = fmaxf(t, __shfl_xor(t, 8, 32));
            const float nm = fmaxf(rmax[v], t);
            const float p  = __expf(z - nm);

            float ps = p;                              // row sum over 16 lanes
            ps += __shfl_xor(ps, 1, 32);
            ps += __shfl_xor(ps, 2, 32);
            ps += __shfl_xor(ps, 4, 32);
            ps += __shfl_xor(ps, 8, 32);

            const float corr = __expf(rmax[v] - nm);
            rsum[v] = rsum[v] * corr + ps;
            rmax[v] = nm;
#pragma unroll
            for (int t4 = 0; t4 < 4; ++t4) o[t4][v] *= corr;

            // stash P (C layout -> LDS) for A-layout reload
            p_lds[wave * 16 * PSTRIDE + (v + hi * 8) * PSTRIDE + lo] = p;
        }
        __syncthreads();

        // O_w += P @ V_w   (A = P 16x16, B = V slice 16x64)
        const float* pb = &p_lds[wave * 16 * PSTRIDE + lo * PSTRIDE + hi * 2];
#pragma unroll
        for (int st = 0; st < 4; ++st) {
            v2f pa = *(const v2f*)(pb + st * 4);
#pragma unroll
            for (int t4 = 0; t4 < 4; ++t4) {
                v2f vb = *(const v2f*)&v_lds[(wave * 64 + t4 * 16 + lo) * VSTRIDE + st * 4 + hi * 2];
                o[t4] = wmma_f32(pa, vb, o[t4]);
            }
        }
    }

    // epilogue: normalize and store this wave's 16 x 64 output slice
    const size_t obase = (size_t)b * SEQ * HID + g * DHEAD + wave * 64;
#pragma unroll
    for (int t4 = 0; t4 < 4; ++t4) {
        const int col = t4 * 16 + lo;
#pragma unroll
        for (int v = 0; v < 8; ++v) {
            const int l = l0 + v + hi * 8;
            OUT[obase + (size_t)l * HID + col] = o[t4][v] / rsum[v];
        }
    }
}

// ---------------------------------------------------------------------------
extern "C" void kernel_launch(void* const* d_in, const int* in_sizes, int n_in,
                              void* d_out, int out_size, void* d_ws, size_t ws_size,
                              hipStream_t stream)
{
    const float* query  = (const float*)d_in[0];
    // d_in[1] key_in, d_in[2] value: unused by the reference computation
    const float* qkv_w  = (const float*)d_in[3];
    const float* qkv_b  = (const float*)d_in[4];
    const float* off_w  = (const float*)d_in[5];
    const float* lin2_w = (const float*)d_in[6];
    const float* lin2_b = (const float*)d_in[7];
    float* out = (float*)d_out;

    float* qkv = (float*)d_ws;                                   // 16384 x 3072 f32
    float* dx  = (float*)((char*)d_ws + (size_t)BATCH * SEQ * QKVC * sizeof(float));

    dim3 gA(QKVC / 64, (BATCH * SEQ) / 128);
    qkv_gemm_kernel<<<gA, 256, 0, stream>>>(query, qkv_w, qkv_b, qkv);

    offset_mlp_kernel<<<(BATCH * SEQ) / 8, 256, 0, stream>>>(query, off_w, lin2_w, lin2_b, dx);

    dim3 gC(SEQ / 16, GROUPS, BATCH);
    attention_kernel<<<gC, 128, 0, stream>>>(qkv, dx, out);
}